// NaiveBayes_65051574665468
// MI455X (gfx1250) — compile-verified
//
#include <hip/hip_runtime.h>
#include <hip/hip_bf16.h>
#include <math.h>

// Problem constants (match reference).
#define EMB        256
#define TWO_D      512          // 2*EMB, the Gaussian dimension
#define KTOT       1024         // folded GEMM K: [x^2 | x]
#define NREL       128
#define BATCH      4096
#define LOG_SQRT_2PI 0.9189385332046727f
#define XSTRIDE    516          // LDS row stride (floats): l16*516%64 = 4*l16 -> 16 distinct banks
#define MROWS      32           // rows per block (2 M-subtiles per wave)

typedef __attribute__((ext_vector_type(2))) float v2f;
typedef __attribute__((ext_vector_type(8))) float v8f;

// ---------------------------------------------------------------------------
// Prep 1: build Wf in WMMA-fragment-ready layout.
//   Folded coefficient matrix W[k'][r]: k'<512 -> -0.5/sigma^2 ; else mu/sigma^2.
//   Fragment layout: Wf[((t*128 + r)*2 + h)*2 + j] = W[4t + 2h + j][r]
//   so a lane (l16=r-r0, lhalf=h) fetches its two B values with ONE b64 load,
//   and a full wave covers one dense 256B region.
// ---------------------------------------------------------------------------
__global__ __launch_bounds__(256)
void nb_prep_w(const float* __restrict__ mus, const float* __restrict__ sigmas,
               float* __restrict__ Wf) {
    int i = blockIdx.x * 256 + threadIdx.x;
    if (i >= TWO_D * NREL) return;
    int k = i / NREL;                  // 0..511
    int r = i - k * NREL;
    float s  = sigmas[r * TWO_D + k];
    float iv = 1.0f / (s * s);
    float av = -0.5f * iv;             // quadratic coefficient
    float bm = mus[r * TWO_D + k] * iv;// linear coefficient
    int t = k >> 2, h = (k >> 1) & 1, j = k & 1;
    Wf[t * 512 + r * 4 + h * 2 + j]            = av;   // k' = k
    Wf[(t + 128) * 512 + r * 4 + h * 2 + j]    = bm;   // k' = k + 512
}

// ---------------------------------------------------------------------------
// Prep 2: per-relation constant C[r] (block reduction over 512 dims).
// ---------------------------------------------------------------------------
__global__ __launch_bounds__(256)
void nb_prep_bias(const float* __restrict__ mus, const float* __restrict__ sigmas,
                  const float* __restrict__ priors, float* __restrict__ C) {
    __shared__ float red[256];
    int r = blockIdx.x;
    float acc = 0.0f;
    for (int k = threadIdx.x; k < TWO_D; k += 256) {
        float s  = sigmas[r * TWO_D + k];
        float mu = mus[r * TWO_D + k];
        float iv = 1.0f / (s * s);
        acc += -0.5f * mu * mu * iv - logf(s);
    }
    red[threadIdx.x] = acc;
    __syncthreads();
    for (int off = 128; off > 0; off >>= 1) {
        if (threadIdx.x < off) red[threadIdx.x] += red[threadIdx.x + off];
        __syncthreads();
    }
    if (threadIdx.x == 0)
        C[r] = red[0] + (float)TWO_D * (priors[r] - LOG_SQRT_2PI);
}

// ---------------------------------------------------------------------------
// Main: 128 blocks x 256 threads. Block owns a 32-row M-tile (staged once in
// LDS); wave w owns 16 relations and TWO 16-row M-subtiles, so each B
// fragment (one b64 load) feeds two v_wmma_f32_16x16x4_f32.
// ---------------------------------------------------------------------------
__global__ __launch_bounds__(256)
void nb_wmma(const float* __restrict__ sbjs, const float* __restrict__ objs,
             const float* __restrict__ Wf, const float* __restrict__ C,
             float* __restrict__ out) {
    __shared__ float xs[MROWS * XSTRIDE];   // 32 rows of x, ~66 KB (<320 KB/WGP)

    const int m0 = blockIdx.x * MROWS;

    // Cooperative, coalesced fill of the x tile (x^2 formed in-register later).
    for (int i = threadIdx.x; i < MROWS * TWO_D; i += 256) {
        int row = i >> 9;                   // /512
        int k   = i & (TWO_D - 1);
        int b   = m0 + row;
        float v = (k < EMB) ? sbjs[b * EMB + k] : objs[b * EMB + (k - EMB)];
        xs[row * XSTRIDE + k] = v;
    }
    __syncthreads();

    const int lane  = threadIdx.x & 31;
    const int wave  = threadIdx.x >> 5;
    const int l16   = lane & 15;            // A-row / B-col / D-col within tile
    const int lhalf = lane >> 4;            // K half (f32 16x16x4 layout)
    const int r0    = wave * 16;

    const float* x0 = &xs[l16 * XSTRIDE + 2 * lhalf];   // M-subtile 0
    const float* x1 = x0 + 16 * XSTRIDE;                // M-subtile 1
    const float* wp = Wf + ((r0 + l16) * 2 + lhalf) * 2;

    v8f acc0 = {};
    v8f acc1 = {};

    // Pass 1: t = 0..127 (k' = 0..511), A = x^2, B = -0.5/sigma^2.
    #pragma unroll 2
    for (int t = 0; t < 128; ++t) {
        v2f b   = *(const v2f*)(wp + t * 512);
        v2f xv0 = *(const v2f*)(x0 + t * 4);
        v2f xv1 = *(const v2f*)(x1 + t * 4);
        v2f a0;  a0.x = xv0.x * xv0.x;  a0.y = xv0.y * xv0.y;
        v2f a1;  a1.x = xv1.x * xv1.x;  a1.y = xv1.y * xv1.y;
        acc0 = __builtin_amdgcn_wmma_f32_16x16x4_f32(
            false, a0, false, b, (short)0, acc0, false, false);
        acc1 = __builtin_amdgcn_wmma_f32_16x16x4_f32(
            false, a1, false, b, (short)0, acc1, false, false);
    }
    // Pass 2: t = 0..127 (k' = 512..1023), A = x, B = mu/sigma^2.
    const float* wp2 = wp + 128 * 512;
    #pragma unroll 2
    for (int t = 0; t < 128; ++t) {
        v2f b  = *(const v2f*)(wp2 + t * 512);
        v2f a0 = *(const v2f*)(x0 + t * 4);
        v2f a1 = *(const v2f*)(x1 + t * 4);
        acc0 = __builtin_amdgcn_wmma_f32_16x16x4_f32(
            false, a0, false, b, (short)0, acc0, false, false);
        acc1 = __builtin_amdgcn_wmma_f32_16x16x4_f32(
            false, a1, false, b, (short)0, acc1, false, false);
    }

    // Epilogue: D layout -> VGPR i: lanes 0-15 row i, lanes 16-31 row i+8.
    float bias = C[r0 + l16];
    float* op0 = out + (m0 + 8 * lhalf) * NREL + r0 + l16;
    float* op1 = op0 + 16 * NREL;
    #pragma unroll
    for (int i = 0; i < 8; ++i) {
        op0[i * NREL] = acc0[i] + bias;
        op1[i * NREL] = acc1[i] + bias;
    }
}

// ---------------------------------------------------------------------------
extern "C" void kernel_launch(void* const* d_in, const int* in_sizes, int n_in,
                              void* d_out, int out_size, void* d_ws, size_t ws_size,
                              hipStream_t stream) {
    const float* sbjs   = (const float*)d_in[0];
    const float* objs   = (const float*)d_in[1];
    const float* mus    = (const float*)d_in[2];
    const float* sigmas = (const float*)d_in[3];
    const float* priors = (const float*)d_in[4];
    float* out = (float*)d_out;

    float* Wf = (float*)d_ws;                // KTOT*NREL floats = 512 KB
    float* C  = Wf + KTOT * NREL;            // NREL floats

    nb_prep_w<<<(TWO_D * NREL + 255) / 256, 256, 0, stream>>>(mus, sigmas, Wf);
    nb_prep_bias<<<NREL, 256, 0, stream>>>(mus, sigmas, priors, C);
    nb_wmma<<<BATCH / MROWS, 256, 0, stream>>>(sbjs, objs, Wf, C, out);
}